// MoERouter_86535001079848
// MI455X (gfx1250) — compile-verified
//
#include <hip/hip_runtime.h>
#include <hip/hip_bf16.h>
#include <math.h>

// MoE router for MI455X (gfx1250, wave32).
// T=16384 tokens, H=4096, E=8 experts, K=2 top-k.
// Bandwidth-bound: 256MB of hidden_states streamed once; K-reduction done with
// v_wmma_f32_16x16x4_f32 (fp32 in/out, matches reference precision).

typedef __attribute__((ext_vector_type(2))) float v2f;
typedef __attribute__((ext_vector_type(8))) float v8f;

#define T_TOK   16384
#define H_DIM   4096
#define E_EXP   8
#define TOPK    2
#define AUXC    0.01f

// d_out layout (floats, concatenated in reference return order):
//   [0, 32768)          routing_weights  (T x 2)
//   [32768, 65536)      selected_experts (T x 2, stored as float-valued ints)
//   [65536, 196608)     router_logits    (T x 8)
//   [196608]            aux_loss
#define OFF_RW   0
#define OFF_SEL  (T_TOK * TOPK)
#define OFF_LOG  (2 * T_TOK * TOPK)
#define OFF_AUX  (2 * T_TOK * TOPK + T_TOK * E_EXP)

// ws layout (floats): [0..7] sum of probs per expert, [8..15] top1 counts.

__global__ void moe_router_init(float* __restrict__ ws) {
    if (threadIdx.x < 2 * E_EXP) ws[threadIdx.x] = 0.0f;
}

__global__ __launch_bounds__(256) void moe_router_main(
    const float* __restrict__ hidden,   // [T, H]
    const float* __restrict__ gate,     // [E, H]
    float* __restrict__ out,
    float* __restrict__ ws)
{
    // Per-wave 16x16 C tile exchange (pad to 17 to dodge bank conflicts),
    // plus block-level partial reductions.
    __shared__ float ldsC[8][16][17];
    __shared__ float ldsP[E_EXP];
    __shared__ float ldsF[E_EXP];

    const int lane = threadIdx.x & 31;
    const int wave = threadIdx.x >> 5;
    const int tile = blockIdx.x * 8 + wave;   // one 16-token tile per wave
    const int rowBase = tile * 16;

    if (threadIdx.x < E_EXP) { ldsP[threadIdx.x] = 0.0f; ldsF[threadIdx.x] = 0.0f; }
    __syncthreads();

    // ---- Phase 1: logits tile via V_WMMA_F32_16X16X4_F32 ------------------
    // A (16x4 f32): lanes 0-15 -> row m, K = {k, k+1}; lanes 16-31 -> row m,
    // K = {k+2, k+3}.  B (4x16): mirrored per-lane pair from gate_w row e.
    const int m     = lane & 15;          // token row in tile / expert column
    const int khalf = (lane >> 4) * 2;    // 0 or 2 -> 8-byte aligned float2
    const int e     = (m < E_EXP) ? m : (m & 7);   // clamp pad columns (unused)

    const float* __restrict__ aptr = hidden + (size_t)(rowBase + m) * H_DIM + khalf;
    const float* __restrict__ bptr = gate   + (size_t)e * H_DIM + khalf;

    v8f acc = {};
    for (int kb = 0; kb < H_DIM; kb += 64) {
        __builtin_prefetch(aptr + 512, 0, 1);   // global_prefetch ~2KB ahead
        #pragma unroll
        for (int kk = 0; kk < 64; kk += 4) {
            v2f a = *(const v2f*)(aptr + kk);   // global_load_b64
            v2f b = *(const v2f*)(bptr + kk);
            acc = __builtin_amdgcn_wmma_f32_16x16x4_f32(
                false, a, false, b, (short)0, acc, false, false);
        }
        aptr += 64; bptr += 64;
    }

    // ---- C tile -> LDS (transpose lane-striped layout to row-per-token) ---
    // C layout: VGPR j, lanes 0-15 -> M=j, lanes 16-31 -> M=j+8; N = lane&15.
    const int crow = (lane >> 4) * 8;
    #pragma unroll
    for (int j = 0; j < 8; ++j)
        ldsC[wave][crow + j][m] = acc[j];
    __syncthreads();

    // ---- Phase 2: per-token softmax / top-2 (lanes 0-15 of each wave) -----
    if (lane < 16) {
        const int t = rowBase + lane;
        float l[E_EXP], p[E_EXP];
        float mx = -INFINITY;
        #pragma unroll
        for (int i = 0; i < E_EXP; ++i) {
            l[i] = ldsC[wave][lane][i];
            mx = fmaxf(mx, l[i]);
        }
        float sum = 0.0f;
        #pragma unroll
        for (int i = 0; i < E_EXP; ++i) { p[i] = __expf(l[i] - mx); sum += p[i]; }
        const float inv = 1.0f / sum;

        #pragma unroll
        for (int i = 0; i < E_EXP; ++i)
            out[OFF_LOG + (size_t)t * E_EXP + i] = l[i];

        // top-2, lowest index wins ties (matches jax.lax.top_k stability)
        int i1 = 0; float p1 = p[0];
        #pragma unroll
        for (int i = 1; i < E_EXP; ++i)
            if (p[i] > p1) { p1 = p[i]; i1 = i; }
        int i2 = -1; float p2 = -INFINITY;
        #pragma unroll
        for (int i = 0; i < E_EXP; ++i)
            if (i != i1 && p[i] > p2) { p2 = p[i]; i2 = i; }

        const float wnorm = 1.0f / (p1 + p2);   // softmax denom cancels
        out[OFF_RW  + (size_t)t * 2    ] = p1 * wnorm;
        out[OFF_RW  + (size_t)t * 2 + 1] = p2 * wnorm;
        out[OFF_SEL + (size_t)t * 2    ] = (float)i1;
        out[OFF_SEL + (size_t)t * 2 + 1] = (float)i2;

        // block partials: P-sum (probs) and top-1 one-hot count
        #pragma unroll
        for (int i = 0; i < E_EXP; ++i)
            atomicAdd(&ldsP[i], p[i] * inv);    // ds_add_f32
        atomicAdd(&ldsF[i1], 1.0f);
    }
    __syncthreads();

    if (threadIdx.x < E_EXP) {
        atomicAdd(&ws[threadIdx.x],        ldsP[threadIdx.x]);  // global_atomic_add_f32
        atomicAdd(&ws[E_EXP + threadIdx.x], ldsF[threadIdx.x]);
    }
}

__global__ void moe_router_aux(const float* __restrict__ ws, float* __restrict__ out) {
    if (threadIdx.x == 0 && blockIdx.x == 0) {
        const float invT = 1.0f / (float)T_TOK;
        float s = 0.0f;
        #pragma unroll
        for (int i = 0; i < E_EXP; ++i)
            s += (ws[E_EXP + i] * invT) * (ws[i] * invT);   // f[e] * P[e]
        out[OFF_AUX] = AUXC * (float)E_EXP * s;
    }
}

extern "C" void kernel_launch(void* const* d_in, const int* in_sizes, int n_in,
                              void* d_out, int out_size, void* d_ws, size_t ws_size,
                              hipStream_t stream) {
    const float* hidden = (const float*)d_in[0];   // [16384, 4096] f32
    const float* gate   = (const float*)d_in[1];   // [8, 4096] f32
    float* out = (float*)d_out;
    float* ws  = (float*)d_ws;

    moe_router_init<<<1, 32, 0, stream>>>(ws);
    // 1024 tiles of 16 tokens; 8 waves per block -> 128 blocks x 256 threads.
    moe_router_main<<<T_TOK / 128, 256, 0, stream>>>(hidden, gate, out, ws);
    moe_router_aux<<<1, 32, 0, stream>>>(ws, out);
}